// CompressorModel_70042326663659
// MI455X (gfx1250) — compile-verified
//
#include <hip/hip_runtime.h>

typedef __attribute__((ext_vector_type(2))) float v2f;
typedef __attribute__((ext_vector_type(4))) float v4f;
typedef __attribute__((ext_vector_type(8))) float v8f;

#define KDIM   768        // 3 * 16 * 16 compressed features per sample
#define NBATCH 32768

// ---------------------------------------------------------------------------
// Kernel 1: fold  (decompress ∘ linear-head)  into one effective weight E[768].
// E[ch, r*8+p, c*8+q] = sum_{P,Q} W[ch*1024 + (r*16+P)*32 + (c*16+Q)]
//                                 * lhs[r*128 + P*8 + p] * rhs[c*128 + q*16 + Q]
// 197K MACs total -> single small block.
// ---------------------------------------------------------------------------
__global__ void build_eff_weights(const float* __restrict__ lhs,
                                  const float* __restrict__ rhs,
                                  const float* __restrict__ W,
                                  float* __restrict__ E) {
    for (int e = threadIdx.x; e < KDIM; e += blockDim.x) {
        const int ch  = e >> 8;          // 0..2
        const int rem = e & 255;
        const int row = rem >> 4;        // r*8 + p  (0..15)
        const int col = rem & 15;        // c*8 + q  (0..15)
        const int r = row >> 3, p = row & 7;
        const int c = col >> 3, q = col & 7;

        const float* lp = lhs + r * 128 + p;          // stride 8 over P
        const float* rp = rhs + c * 128 + q * 16;     // stride 1 over Q
        const float* wp = W + ch * 1024 + (r * 16) * 32 + c * 16;

        float acc = 0.0f;
        #pragma unroll
        for (int P = 0; P < 16; ++P) {
            const float  l    = lp[P * 8];
            const float* wrow = wp + P * 32;
            float s = 0.0f;
            #pragma unroll
            for (int Q = 0; Q < 16; ++Q) s += wrow[Q] * rp[Q];
            acc += l * s;
        }
        E[e] = acc;
    }
}

// ---------------------------------------------------------------------------
// Kernel 2: y[b] = dot(x[b,:], E) + bias  via V_WMMA_F32_16X16X4_F32.
//
// K-chunk of 8 features, with a custom (but A/B-consistent) K-slot mapping:
//   lanes 0-15  (M = lane)    : global_load_b128  x[row][k .. k+3]
//   lanes 16-31 (M = lane-16) : global_load_b128  x[row][k+4 .. k+7]
//   WMMA #0 : A slots = {x[k],x[k+1]} (lo) / {x[k+4],x[k+5]} (hi)
//   WMMA #1 : A slots = {x[k+2],x[k+3]} (lo) / {x[k+6],x[k+7]} (hi)
//   B       : ds_load_b128 of E[k + 4*(lane>=16) .. +3], broadcast over N,
//             split .xy -> WMMA#0, .zw -> WMMA#1.
// Each K-slot pairs x[f] with E[f] for disjoint f => chunk sum exact (fp32).
//
// Every D column equals the running dot; lane0 VGPR0..7 = rows 0..7,
// lane16 VGPR0..7 = rows 8..15 at the end.
// 8 waves/block * 16 rows/wave -> 128 samples/block, 256 blocks.
// Purely HBM-bound: 96 MB of x streamed once (~4.1 us floor @ 23.3 TB/s).
// ---------------------------------------------------------------------------
__global__ void __launch_bounds__(256) gemv_wmma_f32(const float* __restrict__ x,
                                                     const float* __restrict__ E,
                                                     const float* __restrict__ bias,
                                                     float* __restrict__ y) {
    __shared__ float Elds[KDIM];
    for (int i = threadIdx.x; i < KDIM; i += 256) Elds[i] = E[i];
    __syncthreads();

    const int lane    = threadIdx.x & 31;
    const int wave    = threadIdx.x >> 5;
    const int rowBase = blockIdx.x * 128 + wave * 16;
    const int m       = lane & 15;             // batch row within tile
    const int hoff    = (lane >> 4) << 2;      // 0 for lanes 0-15, 4 for 16-31

    const float* xrow = x + (size_t)(rowBase + m) * KDIM + hoff;
    const float* Eh   = Elds + hoff;

    v8f acc = {};
    #pragma unroll 4
    for (int k = 0; k < KDIM; k += 8) {
        v4f xa = *(const v4f*)(xrow + k);      // global_load_b128 (16B aligned)
        v4f eb = *(const v4f*)(Eh + k);        // ds_load_b128, half-wave broadcast

        v2f a0 = { xa.x, xa.y };
        v2f b0 = { eb.x, eb.y };
        // (neg_a, A, neg_b, B, c_mod, C, reuse_a, reuse_b)
        acc = __builtin_amdgcn_wmma_f32_16x16x4_f32(
            false, a0, false, b0, (short)0, acc, false, false);

        v2f a1 = { xa.z, xa.w };
        v2f b1 = { eb.z, eb.w };
        acc = __builtin_amdgcn_wmma_f32_16x16x4_f32(
            false, a1, false, b1, (short)0, acc, false, false);
    }

    const float bs = bias[0];
    if (m == 0) {                              // lane 0 and lane 16 write results
        float* out = y + rowBase + ((lane >> 4) << 3);
        v4f lo = { acc[0] + bs, acc[1] + bs, acc[2] + bs, acc[3] + bs };
        v4f hi = { acc[4] + bs, acc[5] + bs, acc[6] + bs, acc[7] + bs };
        *(v4f*)(out)     = lo;                 // 32B-aligned b128 stores
        *(v4f*)(out + 4) = hi;
    }
}

// ---------------------------------------------------------------------------
extern "C" void kernel_launch(void* const* d_in, const int* in_sizes, int n_in,
                              void* d_out, int out_size, void* d_ws, size_t ws_size,
                              hipStream_t stream) {
    const float* x    = (const float*)d_in[0];   // [32768, 3, 16, 16]
    const float* lhs  = (const float*)d_in[1];   // [2, 16, 8]
    const float* rhs  = (const float*)d_in[2];   // [2, 8, 16]
    const float* W    = (const float*)d_in[3];   // [1, 3072]
    const float* bias = (const float*)d_in[4];   // [1]
    float*       out  = (float*)d_out;           // [32768]
    float*       E    = (float*)d_ws;            // 768 floats of scratch

    build_eff_weights<<<1, 256, 0, stream>>>(lhs, rhs, W, E);
    gemv_wmma_f32<<<NBATCH / 128, 256, 0, stream>>>(x, E, bias, out);
}